// PEG_Shift_171798692026
// MI455X (gfx1250) — compile-verified
//
#include <hip/hip_runtime.h>
#include <cstdint>

// Depthwise 3x3 conv (stride 1, pad 1) with DeepShift quantization:
//   xq = clamp(floor(x * 2^16) / 2^16, -2^15, 2^15 - 1)
//   wq = sign(w) * 2^clip(round(log2|w| + 1e-45), -14, 0)
// Shapes: x (32,384,56,56) f32, w (384,1,3,3) f32, out (32,384,56,56) f32.
//
// Roofline: ~308 MB HBM traffic, 0.69 GFLOP -> AI ~2.25 flop/byte -> pure
// bandwidth problem (~13 us floor at 23.3 TB/s). No WMMA on purpose: a
// depthwise conv has no dense-GEMM mapping (K=1 per group). The CDNA5 path
// that matters is async global->LDS staging (ASYNCcnt) + streaming NT stores.
//
// Layout: one block per (b,c) plane; 224 threads = 7 wave32; 3136 = 224*14
// exactly, so all loops are tail-free. LDS tile is 58x58 with a zero border,
// eliminating every boundary predicate from the 9-tap inner loop. Each thread
// owns one column and walks 14 rows with a rolling 3-row register window
// (3 LDS reads + 9 FMA + 1 NT store per output).

#define DW_H    56
#define DW_W    56
#define DW_HW   (DW_H * DW_W)    // 3136
#define DW_TPB  224              // 7 waves; 3136 / 224 = 14 exact
#define DW_TW   58               // padded tile width/height
#define DW_ROWS_PER_THREAD 14

__device__ __forceinline__ float qfix16(float v) {
    // floor to 2^-16 grid, clamp to [-32768, 32767]
    float q = floorf(v * 65536.0f) * (1.0f / 65536.0f);
    return fminf(fmaxf(q, -32768.0f), 32767.0f);
}

__global__ __launch_bounds__(DW_TPB) void peg_shift_dwconv(
    const float* __restrict__ x,
    const float* __restrict__ wgt,
    float* __restrict__ out,
    int C)
{
    __shared__ __align__(16) float tile[DW_TW * DW_TW];  // 13456 B

    const int plane = blockIdx.x;            // = b*C + c (NCHW)
    const int c     = plane % C;
    const int tid   = threadIdx.x;
    const int r0    = tid / DW_W;            // 0..3  (computed once)
    const int w     = tid - r0 * DW_W;       // 0..55

    // ---- 1) Zero the 1-element border (228 cells). Disjoint from the async
    //         interior writes, so both just need to finish before the barrier.
    if (tid < 228) {
        int idx;
        if      (tid < 58)  idx = tid;                        // top row
        else if (tid < 116) idx = 57 * DW_TW + (tid - 58);    // bottom row
        else if (tid < 172) idx = (tid - 115) * DW_TW;        // left col
        else                idx = (tid - 171) * DW_TW + 57;   // right col
        tile[idx] = 0.0f;
    }

    // ---- 2) Async global->LDS copy of the interior (rows r0+4k, column w).
    //         Lanes 0..31 cover consecutive columns -> 128 B contiguous per
    //         wave per instruction; 14 outstanding << ASYNCcnt max 63.
    {
        const float* g  = x + (size_t)plane * DW_HW + (size_t)(r0 * DW_W + w);
        uint32_t     la = (uint32_t)(uintptr_t)(&tile[(r0 + 1) * DW_TW + (w + 1)]);
#pragma unroll
        for (int k = 0; k < DW_ROWS_PER_THREAD; ++k) {
            asm volatile("global_load_async_to_lds_b32 %0, %1, off"
                         :: "v"(la), "v"(g)
                         : "memory");
            g  += 4 * DW_W;               // 4 rows ahead in global
            la += 4 * DW_TW * 4;          // 4 rows ahead in padded LDS (bytes)
        }
    }

    // ---- 3) Overlap: power-of-two quantize this channel's 9 weights.
    float wq[9];
    {
        const float* wp = wgt + c * 9;
#pragma unroll
        for (int i = 0; i < 9; ++i) {
            float ww = wp[i];
            float sh = rintf(log2f(fabsf(ww) + 1e-45f));  // round-half-even
            sh = fminf(fmaxf(sh, -14.0f), 0.0f);
            float s  = (ww > 0.0f) ? 1.0f : ((ww < 0.0f) ? -1.0f : 0.0f);
            wq[i] = s * exp2f(sh);                         // exact: int exponent
        }
    }

    asm volatile("s_wait_asynccnt 0x0" ::: "memory");
    __syncthreads();

    // ---- 4) Fixed-point quantize activations ONCE in LDS (border is 0 and
    //         qfix16(0) == 0, so only the interior needs touching).
    {
        float* p = &tile[(r0 + 1) * DW_TW + (w + 1)];
#pragma unroll
        for (int k = 0; k < DW_ROWS_PER_THREAD; ++k) {
            *p = qfix16(*p);
            p += 4 * DW_TW;
        }
    }
    __syncthreads();

    // ---- 5) 3x3 conv: thread = column w, rows h = r0*14 .. r0*14+13.
    //         Rolling 3-row window: 3 LDS reads + 9 FMA + 1 NT store / output.
    {
        const int hbase = r0 * DW_ROWS_PER_THREAD;
        // Output o[h][w] needs padded tile rows h..h+2, cols w..w+2.
        const float* ta = &tile[hbase * DW_TW + w];
        float a0 = ta[0],          a1 = ta[1],          a2 = ta[2];
        float b0 = ta[DW_TW],      b1 = ta[DW_TW + 1],  b2 = ta[DW_TW + 2];
        const float* tc = ta + 2 * DW_TW;
        float* op = out + (size_t)plane * DW_HW + (size_t)hbase * DW_W + w;

#pragma unroll
        for (int k = 0; k < DW_ROWS_PER_THREAD; ++k) {
            float c0 = tc[0], c1 = tc[1], c2 = tc[2];
            float acc;
            acc = wq[0] * a0;
            acc = fmaf(wq[1], a1, acc);
            acc = fmaf(wq[2], a2, acc);
            acc = fmaf(wq[3], b0, acc);
            acc = fmaf(wq[4], b1, acc);
            acc = fmaf(wq[5], b2, acc);
            acc = fmaf(wq[6], c0, acc);
            acc = fmaf(wq[7], c1, acc);
            acc = fmaf(wq[8], c2, acc);
            __builtin_nontemporal_store(acc, op);  // streamed, no reuse

            a0 = b0; a1 = b1; a2 = b2;             // unrolled -> pure renaming
            b0 = c0; b1 = c1; b2 = c2;
            tc += DW_TW;
            op += DW_W;
        }
    }
}

extern "C" void kernel_launch(void* const* d_in, const int* in_sizes, int n_in,
                              void* d_out, int out_size, void* d_ws, size_t ws_size,
                              hipStream_t stream) {
    const float* x   = (const float*)d_in[0];
    const float* wgt = (const float*)d_in[1];
    float*       out = (float*)d_out;

    const int planes = in_sizes[0] / DW_HW;  // B*C = 12288
    const int C      = in_sizes[1] / 9;      // 384

    peg_shift_dwconv<<<planes, DW_TPB, 0, stream>>>(x, wgt, out, C);
}